// GCN_81037442940969
// MI455X (gfx1250) — compile-verified
//
#include <hip/hip_runtime.h>
#include <hip/hip_bf16.h>

// GCNConv: out = scatter-norm( (x @ W) ), N=100000, K=1024, H=128, E=1.6M
//
// MI455X plan:
//  * GEMM via V_WMMA_F32_16X16X4_F32 (exact fp32, matrix pipe). 460MB HBM
//    traffic -> ~20us floor at 23.3 TB/s.
//  * x stream (410MB, HBM-bound) double-buffered into LDS with
//    GLOBAL_LOAD_ASYNC_TO_LDS_B128 (ASYNCcnt): chunk k+1 fetch overlaps WMMA
//    on chunk k. W chunk (L2-hot) async-loaded with a k-paired swizzle via
//    per-lane GLOBAL_LOAD_ASYNC_TO_LDS_B32 so every B fragment is one
//    aligned ds_load_b64 (no VGPR re-pairing movs). W issued before x-next,
//    async loads complete in order => s_wait_asynccnt<=4 releases W while x
//    stays in flight.
//  * B fragments batched per kk-step (8 independent ds_load_b64 up front,
//    then 8 WMMAs) so the scheduler drains DScnt incrementally instead of
//    waiting dscnt==0 before every pair.
//  * Bank-conflict-free LDS: W pair-row stride 288 floats (halves' bank sets
//    offset by 32 -> disjoint); x row stride 36 floats (36*m mod 64 distinct,
//    +2 half offset unsolvable mod 64).
//  * h' (=h*dinv[row], 51MB) and out (51MB) both fit the 192MB L2, so the
//    1.6GB edge gather/atomic-scatter stream is L2-resident.
//  * out[c] = dinv[c] * (h'[c] + sum_{e: col=c} h'[row_e])  (self-loop folded
//    into the accumulator init; per-edge multiply eliminated).

typedef float v2f __attribute__((ext_vector_type(2)));
typedef float v8f __attribute__((ext_vector_type(8)));

#define GCN_K    1024
#define GCN_H    128
#define GCN_KC   32          // K-chunk staged in LDS per iteration
#define GCN_WP2  288         // floats per W pair-row (2*128 data + pad)
#define GCN_XP   36          // padded x row stride (floats)
#define GCN_XBUF (8 * 16 * GCN_XP * 4)   // bytes per x buffer (all 8 waves)

// Async copies: each lane copies 16B (b128) or 4B (b32) global -> LDS.
__device__ __forceinline__ void gcn_async_b128(unsigned lds_off, const float* g) {
    asm volatile("global_load_async_to_lds_b128 %0, %1, off"
                 :: "v"(lds_off), "v"(g) : "memory");
}
__device__ __forceinline__ void gcn_async_b32(unsigned lds_off, const float* g) {
    asm volatile("global_load_async_to_lds_b32 %0, %1, off"
                 :: "v"(lds_off), "v"(g) : "memory");
}

// ---------------- degree / normalization ----------------

__global__ void gcn_deg_init(float* __restrict__ deg, int n) {
    int i = blockIdx.x * blockDim.x + threadIdx.x;
    if (i < n) deg[i] = 1.0f;                 // self-loop
}

__global__ void gcn_deg_count(const int* __restrict__ col, float* __restrict__ deg, int e) {
    int i = blockIdx.x * blockDim.x + threadIdx.x;
    int stride = gridDim.x * blockDim.x;
    for (; i < e; i += stride) atomicAdd(&deg[col[i]], 1.0f);
}

__global__ void gcn_rsqrt(const float* __restrict__ deg, float* __restrict__ dinv, int n) {
    int i = blockIdx.x * blockDim.x + threadIdx.x;
    if (i < n) dinv[i] = rsqrtf(deg[i]);
}

// ---------------- WMMA GEMM: h' = (x @ W) * dinv[row] ----------------
// One wave computes a 16(M) x 128(N) strip: 8 accumulators of 16x16.
// Grid: ceil((n/16)/8) blocks of 256 threads (8 waves).

__global__ __launch_bounds__(256) void gcn_gemm_wmma(
    const float* __restrict__ x, const float* __restrict__ W,
    const float* __restrict__ dinv, float* __restrict__ h,
    float* __restrict__ out, int n)
{
    // W element (k,n) lives at sW[k>>1][2n + (k&1)] -> B frag = one b64.
    __shared__ __align__(16) float sW[GCN_KC / 2][GCN_WP2];     // 16 x 288 (18.4KB)
    __shared__ __align__(16) float sX[2][8][16][GCN_XP];        // double buffer (2x18.4KB)

    const int wave = threadIdx.x >> 5;
    const int lane = threadIdx.x & 31;
    const int row0 = (blockIdx.x * 8 + wave) * 16;
    const bool active = (row0 < n);

    const int m  = lane & 15;          // M (A) / N-in-tile (B,C,D)
    const int kh = (lane >> 4) * 2;    // K sub-pair selected by wave half

    const unsigned sW_base = (unsigned)(uintptr_t)&sW[0][0];
    const unsigned sX_base = (unsigned)(uintptr_t)&sX[0][0][0][0];

    // Per-lane pieces of the x async copies (lane covers 16B of a row).
    const int xr = lane >> 3;                 // 0..3: row within group of 4
    const int xc = (lane & 7) * 4;            // float col within 32-wide chunk

    // Prologue: prefetch x chunk 0 into buffer 0 (4 rows per B128 issue).
    if (active) {
        const float* gx = x + (size_t)row0 * GCN_K;
        unsigned lx = sX_base + (unsigned)(wave * 16 * GCN_XP * 4);
        #pragma unroll
        for (int j = 0; j < 4; ++j) {
            int rr = j * 4 + xr;
            gcn_async_b128(lx + (unsigned)(rr * GCN_XP * 4 + (lane & 7) * 16),
                           gx + (size_t)rr * GCN_K + xc);
        }
    }

    v8f acc[8] = {};
    int buf = 0;

    for (int k0 = 0; k0 < GCN_K; k0 += GCN_KC) {
        __syncthreads();   // nobody still reads sW / the buffer we overwrite

        // Async-load this chunk's W (32x128) with the k-paired swizzle:
        // element i -> k=i>>7, n=i&127 -> LDS (k>>1)*WP2 + 2n + (k&1).
        // 16 b32 issues per thread; global side coalesced along n.
        #pragma unroll
        for (int j = 0; j < 16; ++j) {
            int i = j * 256 + threadIdx.x;
            int k = i >> 7, nn = i & 127;
            gcn_async_b32(sW_base + (unsigned)(((k >> 1) * GCN_WP2 + 2 * nn + (k & 1)) * 4),
                          W + (size_t)(k0 + k) * GCN_H + nn);
        }

        // Prefetch NEXT chunk's x tile into the alternate buffer.
        const bool pre = active && (k0 + GCN_KC < GCN_K);
        if (pre) {
            const float* gx = x + (size_t)row0 * GCN_K + (k0 + GCN_KC);
            unsigned lx = sX_base + (unsigned)((buf ^ 1) * GCN_XBUF + wave * 16 * GCN_XP * 4);
            #pragma unroll
            for (int j = 0; j < 4; ++j) {
                int rr = j * 4 + xr;
                gcn_async_b128(lx + (unsigned)(rr * GCN_XP * 4 + (lane & 7) * 16),
                               gx + (size_t)rr * GCN_K + xc);
            }
        }

        // W issued before x-next; async loads complete in order, so <=4
        // outstanding means W (and everything older, incl. current x) landed.
        if (pre) asm volatile("s_wait_asynccnt 0x4" ::: "memory");
        else     asm volatile("s_wait_asynccnt 0x0" ::: "memory");
        __syncthreads();

        if (active) {
            #pragma unroll
            for (int kk = 0; kk < GCN_KC; kk += 4) {
                // A 16x4 f32: lane m=L%16, K pair = (L/16)*2 + vgpr.
                // Adjacent floats -> single aligned b64.
                v2f a = *(const v2f*)&sX[buf][wave][m][kk + kh];
                const int p = (kk + kh) >> 1;   // W pair-row

                // Batch all 8 B fragments first (independent ds_load_b64s)
                // so the scheduler can drain DScnt incrementally under the
                // WMMA issues instead of s_wait_dscnt 0 per pair.
                v2f b[8];
                #pragma unroll
                for (int t = 0; t < 8; ++t)
                    b[t] = *(const v2f*)&sW[p][2 * (t * 16 + m)];

                #pragma unroll
                for (int t = 0; t < 8; ++t)
                    acc[t] = __builtin_amdgcn_wmma_f32_16x16x4_f32(
                        /*neg_a=*/false, a, /*neg_b=*/false, b[t],
                        /*c_mod=*/(short)0, acc[t],
                        /*reuse_a=*/false, /*reuse_b=*/false);
            }
        }
        buf ^= 1;
    }

    if (active) {
        // C/D layout: VGPR r -> row r (lanes 0-15) or r+8 (lanes 16-31),
        // column = tile*16 + lane%16. Scale by dinv[row]; write h' and
        // initialize the output accumulator with the self-loop term.
        const int half = (lane >> 4) * 8;
        #pragma unroll
        for (int r = 0; r < 8; ++r) {
            int row = row0 + half + r;
            float s = dinv[row];
            #pragma unroll
            for (int t = 0; t < 8; ++t) {
                float v = acc[t][r] * s;
                size_t idx = (size_t)row * GCN_H + t * 16 + m;
                h[idx]   = v;
                out[idx] = v;
            }
        }
    }
}

// ---------------- edge scatter: out[col] += h'[row] ----------------
// One wave per edge; lane handles 4 contiguous feats (float4 gather, L2-hot).

__global__ void gcn_scatter(const int* __restrict__ row, const int* __restrict__ col,
                            const float* __restrict__ h, float* __restrict__ out, int e)
{
    int gid  = blockIdx.x * blockDim.x + threadIdx.x;
    int wid  = gid >> 5;
    int lane = gid & 31;
    int wstride = (gridDim.x * blockDim.x) >> 5;
    for (int i = wid; i < e; i += wstride) {
        int r = row[i], c = col[i];
        const float4 v = *(const float4*)(h + (size_t)r * GCN_H + lane * 4);
        float* o = out + (size_t)c * GCN_H + lane * 4;
        atomicAdd(o + 0, v.x);
        atomicAdd(o + 1, v.y);
        atomicAdd(o + 2, v.z);
        atomicAdd(o + 3, v.w);
    }
}

// ---------------- final scale: out[c] *= dinv[c] ----------------

__global__ void gcn_scale(float* __restrict__ out, const float* __restrict__ dinv, int total) {
    int i = blockIdx.x * blockDim.x + threadIdx.x;
    if (i < total) out[i] *= dinv[i >> 7];     // H = 128
}

extern "C" void kernel_launch(void* const* d_in, const int* in_sizes, int n_in,
                              void* d_out, int out_size, void* d_ws, size_t ws_size,
                              hipStream_t stream) {
    const float* x  = (const float*)d_in[0];
    const float* W  = (const float*)d_in[1];
    const int*   ei = (const int*)d_in[2];        // edge_index [2, E] flat

    const int n = in_sizes[0] / GCN_K;            // 100000
    const int e = in_sizes[2] / 2;                // 1600000
    const int* erow = ei;
    const int* ecol = ei + e;

    float* ws   = (float*)d_ws;
    float* deg  = ws;                             // n floats
    float* dinv = ws + n;                         // n floats
    float* h    = ws + 2 * (size_t)n;             // n*128 floats (~51 MB)
    float* out  = (float*)d_out;

    // 1) deg = 1 (self loop) + edge counts on destinations
    gcn_deg_init<<<(n + 255) / 256, 256, 0, stream>>>(deg, n);
    gcn_deg_count<<<2048, 256, 0, stream>>>(ecol, deg, e);
    // 2) dinv = rsqrt(deg)
    gcn_rsqrt<<<(n + 255) / 256, 256, 0, stream>>>(deg, dinv, n);
    // 3) h' = (x @ W) * dinv[row]   (WMMA + async double-buffered x stream)
    int tiles  = n / 16;                          // 6250
    int blocks = (tiles + 7) / 8;                 // 782
    gcn_gemm_wmma<<<blocks, 256, 0, stream>>>(x, W, dinv, h, out, n);
    // 4) out[col] += h'[row] over all edges (L2-resident atomics)
    gcn_scatter<<<4096, 256, 0, stream>>>(erow, ecol, h, out, e);
    // 5) out[c] *= dinv[c]
    int total = n * GCN_H;
    gcn_scale<<<(total + 255) / 256, 256, 0, stream>>>(out, dinv, total);
}